// DCSCNet_19069654794620
// MI455X (gfx1250) — compile-verified
//
#include <hip/hip_runtime.h>

typedef float v2f __attribute__((ext_vector_type(2)));
typedef float v8f __attribute__((ext_vector_type(8)));
typedef unsigned int u32x4 __attribute__((ext_vector_type(4)));
typedef int i32x4 __attribute__((ext_vector_type(4)));
typedef int i32x8 __attribute__((ext_vector_type(8)));

#define LSEQ 318        // pooled sequence length
#define NB   128        // batch
#define XL   5087       // raw input length
#define KW   1700       // stem filter width

#if defined(__has_builtin)
#if __has_builtin(__builtin_amdgcn_tensor_load_to_lds) && \
    __has_builtin(__builtin_amdgcn_s_wait_tensorcnt)
#define USE_TDM 1
#endif
#endif

// ds_swizzle group-of-32 XOR mode: offset = (xor_mask<<10) | and_mask(0x1F)
#define SWIZ_MAX(v, imm)                                                     \
    fmaxf((v), __int_as_float(__builtin_amdgcn_ds_swizzle(__float_as_int(v), \
                                                          ((imm) << 10) | 0x1F)))

// ---------------------------------------------------------------------------
// Stem: conv1d(x, filter0, pad=850) + b0, fused maxpool16.
// Block = 8 waves = 8 consecutive pool groups of one batch row. The block's
// x-window (1827 floats) is staged into LDS (TDM tensor_load_to_lds when the
// builtin exists; right edge zero-filled by TDM OOB semantics, left edge
// pre-zeroed with DS stores), so the inner loop is branch-free:
// global b64 (filter) + 2 ds loads + WMMA.
// Grid: (40, 128); waves with g >= 318 compute on in-window data, stores gated.
// ---------------------------------------------------------------------------
__global__ void stem_conv_pool_wmma(const float* __restrict__ x,
                                    const float* __restrict__ w,
                                    const float* __restrict__ b0,
                                    float* __restrict__ out /*[128,16,318]*/) {
    __shared__ float sx[1832];
    const int widx = threadIdx.x >> 5;
    const int lane = threadIdx.x & 31;
    const int g0 = blockIdx.x << 3;
    const int b  = blockIdx.y;
    const int g  = g0 + widx;

    // window covers x[b, 16*g0-850 .. 16*g0+976] (1827 floats), zero padded
    const int wbase = 16 * g0 - 850;
    const float* xb = x + (size_t)b * XL;
#ifdef USE_TDM
    const int lead = (wbase < 0) ? -wbase : 0;
    for (int i = threadIdx.x; i < lead; i += 256) sx[i] = 0.0f;  // left pad
    if (widx == 0) {
        const unsigned lds_base = (unsigned)(size_t)(&sx[lead]);
        const unsigned long long gaddr =
            (unsigned long long)(const void*)(xb + (wbase + lead));
        const unsigned tile0 = 1827u - (unsigned)lead;           // tile_dim0
        const unsigned tdim0 = (unsigned)(XL - (wbase + lead));  // row remaining
        u32x4 g0d;
        g0d.x = 1u;                                          // count=1, valid
        g0d.y = lds_base;                                    // lds_addr (bytes)
        g0d.z = (unsigned)(gaddr & 0xffffffffull);           // global_addr lo
        g0d.w = (unsigned)((gaddr >> 32) & 0x01ffffffull) |  // global_addr hi
                (2u << 30);                                  // type=2 (image)
        i32x8 g1d;
        g1d[0] = 0x00020000;                                 // data_size=4B
        g1d[1] = (int)((tdim0 & 0xffffu) << 16);             // tensor_dim0 lo16
        g1d[2] = (int)((tdim0 >> 16) | (1u << 16));          // dim0 hi16, dim1=1
        g1d[3] = (int)(tile0 << 16);                         // tile_dim0
        g1d[4] = 0;                                          // tile_dim1/2 = 0
        g1d[5] = XL;                                         // dim0 stride lo
        g1d[6] = 0;
        g1d[7] = 0;
        i32x4 z4 = {0, 0, 0, 0};
#if __clang_major__ >= 23
        i32x8 z8 = {0, 0, 0, 0, 0, 0, 0, 0};
        __builtin_amdgcn_tensor_load_to_lds(g0d, g1d, z4, z4, z8, 0);
#else
        __builtin_amdgcn_tensor_load_to_lds(g0d, g1d, z4, z4, 0);
#endif
        __builtin_amdgcn_s_wait_tensorcnt(0);
    }
    __syncthreads();
#else
    for (int i = threadIdx.x; i < 1827; i += 256) {
        const int p = wbase + i;
        sx[i] = ((unsigned)p < (unsigned)XL) ? xb[p] : 0.0f;
    }
    __syncthreads();
#endif

    const int m  = lane & 15;      // A row (channel) / B col (position)
    const int hi = lane >> 4;      // K-pair selector (16x16x4 f32 layout)
    const int sbase = 16 * widx + m;
    const float* wrow = w + m * KW;
    v8f acc = {};
#pragma unroll 5
    for (int k0 = 0; k0 < KW; k0 += 4) {
        const int kk = k0 + 2 * hi;
        v2f av = *(const v2f*)(wrow + kk);       // filter[m][kk, kk+1]
        v2f bv;
        bv.x = sx[sbase + kk];
        bv.y = sx[sbase + kk + 1];
        acc = __builtin_amdgcn_wmma_f32_16x16x4_f32(false, av, false, bv,
                                                    (short)0, acc, false, false);
    }
    // maxpool over N (16 lanes per half-group); bias is max-invariant shift
#pragma unroll
    for (int r = 0; r < 8; ++r) {
        float v = acc[r];
        v = SWIZ_MAX(v, 1);
        v = SWIZ_MAX(v, 2);
        v = SWIZ_MAX(v, 4);
        v = SWIZ_MAX(v, 8);
        acc[r] = v;
    }
    if (m == 0 && g < LSEQ) {
#pragma unroll
        for (int r = 0; r < 8; ++r) {
            const int ch = hi * 8 + r;
            out[((size_t)b * 16 + ch) * LSEQ + g] = acc[r] + b0[ch];
        }
    }
}

// ---------------------------------------------------------------------------
// Generic k=3 conv1d (pad 1) as WMMA GEMM, branchless edge handling
// (clamped-address load + cndmask select; no EXEC writes in the hot loop).
//   y[b,co,l] = sum_{ci,t} w[co,ci,t] * in[b, ci_off+ci, l+t-1]
// ---------------------------------------------------------------------------
__global__ void conv_k3_wmma(const float* __restrict__ in, int in_cstride,
                             int ci_off, int cin,
                             const float* __restrict__ w, int cout,
                             float* __restrict__ out /*[128,cout,318]*/) {
    const int wave = (blockIdx.x << 3) + (threadIdx.x >> 5);
    const int lane = threadIdx.x & 31;
    const int mtiles = cout >> 4;
    const int mt = wave % mtiles;
    const int t2 = wave / mtiles;
    const int lt = t2 % 20;
    const int b  = t2 / 20;
    const int m  = lane & 15;
    const int hi = lane >> 4;
    const int l0 = lt * 16;
    const int Ktot = cin * 3;
    const float* wrow = w + (size_t)(mt * 16 + m) * Ktot;
    const float* inb  = in + ((size_t)b * in_cstride + ci_off) * LSEQ;
    const int pos = l0 + m;   // B column
    v8f acc = {};
#pragma unroll 4
    for (int k0 = 0; k0 < Ktot; k0 += 4) {
        const int kk = k0 + 2 * hi;
        v2f av = *(const v2f*)(wrow + kk);
        v2f bv;
        {
            int k = kk;
            int ci = k / 3; int t = k - 3 * ci;
            unsigned pu = (unsigned)(pos + t - 1);
            float vx = inb[ci * LSEQ + (pu < 318u ? pu : 317u)];
            bv.x = (pu < 318u) ? vx : 0.0f;
            k = kk + 1;
            ci = k / 3; t = k - 3 * ci;
            pu = (unsigned)(pos + t - 1);
            vx = inb[ci * LSEQ + (pu < 318u ? pu : 317u)];
            bv.y = (pu < 318u) ? vx : 0.0f;
        }
        acc = __builtin_amdgcn_wmma_f32_16x16x4_f32(false, av, false, bv,
                                                    (short)0, acc, false, false);
    }
    if (pos < LSEQ) {
#pragma unroll
        for (int r = 0; r < 8; ++r) {
            out[((size_t)b * cout + mt * 16 + r + 8 * hi) * LSEQ + pos] = acc[r];
        }
    }
}

// ---------------------------------------------------------------------------
// FC1: [128,35616] @ [128,35616]^T + bias, ReLU. 64 waves total.
// ---------------------------------------------------------------------------
__global__ void fc1_wmma(const float* __restrict__ z, const float* __restrict__ W,
                         const float* __restrict__ bias, float* __restrict__ out) {
    const int wave = (blockIdx.x << 3) + (threadIdx.x >> 5);
    const int lane = threadIdx.x & 31;
    const int mt = wave & 7;     // batch tile
    const int nt = wave >> 3;    // out-feature tile
    const int m  = lane & 15;
    const int hi = lane >> 4;
    const float* zrow = z + (size_t)(mt * 16 + m) * 35616;
    const float* wrow = W + (size_t)(nt * 16 + m) * 35616;
    v8f acc = {};
#pragma unroll 2
    for (int k0 = 0; k0 < 35616; k0 += 4) {
        const int kk = k0 + 2 * hi;
        if ((k0 & 127) == 0) {
            __builtin_prefetch(zrow + kk + 512, 0, 1);
            __builtin_prefetch(wrow + kk + 512, 0, 1);
        }
        v2f av = *(const v2f*)(zrow + kk);
        v2f bv = *(const v2f*)(wrow + kk);
        acc = __builtin_amdgcn_wmma_f32_16x16x4_f32(false, av, false, bv,
                                                    (short)0, acc, false, false);
    }
#pragma unroll
    for (int r = 0; r < 8; ++r) {
        const int row = mt * 16 + r + 8 * hi;
        const int col = nt * 16 + m;
        out[row * 128 + col] = fmaxf(acc[r] + bias[col], 0.0f);
    }
}

// --------------------------- BatchNorm (training stats) ---------------------
__global__ void bn_stats(const float* __restrict__ in, int C,
                         const float* __restrict__ gamma,
                         const float* __restrict__ beta,
                         float* __restrict__ scale, float* __restrict__ shift) {
    const int c = blockIdx.x;
    __shared__ float s1[256], s2[256];
    float a = 0.f, sq = 0.f;
    for (int i = threadIdx.x; i < NB * LSEQ; i += 256) {
        const int bb = i / LSEQ, l = i - bb * LSEQ;
        const float v = in[((size_t)bb * C + c) * LSEQ + l];
        a += v; sq += v * v;
    }
    s1[threadIdx.x] = a; s2[threadIdx.x] = sq;
    __syncthreads();
    for (int st = 128; st > 0; st >>= 1) {
        if (threadIdx.x < st) {
            s1[threadIdx.x] += s1[threadIdx.x + st];
            s2[threadIdx.x] += s2[threadIdx.x + st];
        }
        __syncthreads();
    }
    if (threadIdx.x == 0) {
        const float n = (float)(NB * LSEQ);
        const float mean = s1[0] / n;
        const float var = s2[0] / n - mean * mean;
        const float sc = gamma[c] * rsqrtf(var + 1e-5f);
        scale[c] = sc;
        shift[c] = beta[c] - mean * sc;
    }
}

__global__ void bn_apply(float* __restrict__ dst, const float* __restrict__ src,
                         const float* __restrict__ scale,
                         const float* __restrict__ shift, int C, int relu) {
    const size_t idx = (size_t)blockIdx.x * blockDim.x + threadIdx.x;
    const size_t total = (size_t)NB * C * LSEQ;
    if (idx >= total) return;
    const int c = (int)((idx / LSEQ) % C);
    float v = src[idx] * scale[c] + shift[c];
    dst[idx] = relu ? fmaxf(v, 0.f) : v;
}

// --------------------------- ISTA elementwise glue --------------------------
__global__ void build_f3(float* __restrict__ out, const float* __restrict__ fi,
                         const float* __restrict__ tconv, const float* __restrict__ bv,
                         const float* __restrict__ cptr, int cin, int C) {
    const size_t idx = (size_t)blockIdx.x * blockDim.x + threadIdx.x;
    const size_t total = (size_t)NB * C * LSEQ;
    if (idx >= total) return;
    const int l = (int)(idx % LSEQ);
    const size_t t = idx / LSEQ;
    const int ch = (int)(t % C);
    const int b = (int)(t / C);
    const float c = cptr[0];
    const float v = (ch < cin) ? fi[((size_t)b * cin + ch) * LSEQ + l]
                               : tconv[((size_t)b * 32 + (ch - cin)) * LSEQ + l];
    out[idx] = c * v + bv[ch];
}

// out[b,ch,l] = A[b,ch,l] + Bsrc[b, ch(of BC), l]   (first C channels of Bsrc)
__global__ void add_sliced(float* __restrict__ out, const float* __restrict__ A,
                           const float* __restrict__ Bsrc, int BC, int C) {
    const size_t idx = (size_t)blockIdx.x * blockDim.x + threadIdx.x;
    const size_t total = (size_t)NB * C * LSEQ;
    if (idx >= total) return;
    const int l = (int)(idx % LSEQ);
    const size_t t = idx / LSEQ;
    const int ch = (int)(t % C);
    const int b = (int)(t / C);
    out[idx] = A[idx] + Bsrc[((size_t)b * BC + ch) * LSEQ + l];
}

// f7 = feats[i+1] - c*concat(T2 - feats[i], T3 - T4) + bv
__global__ void combine_f7(float* __restrict__ out, const float* __restrict__ fnext,
                           const float* __restrict__ fi, const float* __restrict__ t2,
                           const float* __restrict__ t3, const float* __restrict__ t4,
                           const float* __restrict__ bv, const float* __restrict__ cptr,
                           int cin, int C) {
    const size_t idx = (size_t)blockIdx.x * blockDim.x + threadIdx.x;
    const size_t total = (size_t)NB * C * LSEQ;
    if (idx >= total) return;
    const int l = (int)(idx % LSEQ);
    const size_t t = idx / LSEQ;
    const int ch = (int)(t % C);
    const int b = (int)(t / C);
    const float c = cptr[0];
    float diff;
    if (ch < cin)
        diff = t2[((size_t)b * cin + ch) * LSEQ + l] - fi[((size_t)b * cin + ch) * LSEQ + l];
    else
        diff = t3[((size_t)b * 32 + ch - cin) * LSEQ + l] - t4[((size_t)b * 32 + ch - cin) * LSEQ + l];
    out[idx] = fnext[idx] - c * diff + bv[ch];
}

// wt[o][i][t] = w[i][o][2-t]  with w:[32, coutT, 3]
__global__ void wtrans(const float* __restrict__ w, float* __restrict__ wt, int coutT) {
    const int idx = blockIdx.x * blockDim.x + threadIdx.x;
    const int total = coutT * 32 * 3;
    if (idx >= total) return;
    const int t = idx % 3;
    const int r = idx / 3;
    const int i = r % 32;
    const int o = r / 32;
    wt[idx] = w[(i * coutT + o) * 3 + (2 - t)];
}

// ------------------------------- eSE ----------------------------------------
__global__ void ese_mean(const float* __restrict__ z, float* __restrict__ s) {
    const int idx = blockIdx.x * blockDim.x + threadIdx.x;   // b*112 + c
    if (idx >= NB * 112) return;
    const float* p = z + (size_t)idx * LSEQ;
    float a = 0.f;
    for (int l = 0; l < LSEQ; ++l) a += p[l];
    s[idx] = a / (float)LSEQ;
}

__global__ void ese_gate(const float* __restrict__ s, const float* __restrict__ W,
                         const float* __restrict__ bias, float* __restrict__ a) {
    const int idx = blockIdx.x * blockDim.x + threadIdx.x;
    if (idx >= NB * 112) return;
    const int c = idx % 112;
    const int b = idx / 112;
    const float* sr = s + b * 112;
    const float* wr = W + c * 112;
    float acc = bias[c];
    for (int k = 0; k < 112; ++k) acc += sr[k] * wr[k];
    acc += 3.f;
    a[idx] = fminf(fmaxf(acc, 0.f), 6.f) * (1.f / 6.f);
}

__global__ void ese_scale(const float* __restrict__ z, const float* __restrict__ a,
                          float* __restrict__ out) {
    const size_t idx = (size_t)blockIdx.x * blockDim.x + threadIdx.x;
    const size_t total = (size_t)NB * 112 * LSEQ;
    if (idx >= total) return;
    out[idx] = z[idx] * a[idx / LSEQ];
}

// ------------------------- FC2 + log_softmax --------------------------------
__global__ void fc2_logsoftmax(const float* __restrict__ h, const float* __restrict__ W,
                               const float* __restrict__ bias, float* __restrict__ out) {
    const int b = blockIdx.x;
    __shared__ float logits[10];
    __shared__ float red[2];
    const int t = threadIdx.x;
    if (t < 10) {
        const float* hr = h + b * 128;
        const float* wr = W + t * 128;
        float acc = bias[t];
        for (int k = 0; k < 128; ++k) acc += hr[k] * wr[k];
        logits[t] = acc;
    }
    __syncthreads();
    if (t == 0) {
        float mx = logits[0];
        for (int i = 1; i < 10; ++i) mx = fmaxf(mx, logits[i]);
        float se = 0.f;
        for (int i = 0; i < 10; ++i) se += __expf(logits[i] - mx);
        red[0] = mx;
        red[1] = __logf(se);
    }
    __syncthreads();
    if (t < 10) {
        const float lg = logits[t];
        out[b * 10 + t] = lg - red[0] - red[1];   // log_softmax
        out[1280 + b * 10 + t] = lg;              // logits
    }
}

// ---------------------------------------------------------------------------
static inline int eb(size_t total) { return (int)((total + 255) / 256); }

extern "C" void kernel_launch(void* const* d_in, const int* in_sizes, int n_in,
                              void* d_out, int out_size, void* d_ws, size_t ws_size,
                              hipStream_t stream) {
    const float* x      = (const float*)d_in[0];
    const float* f0     = (const float*)d_in[1];
    const float* b0     = (const float*)d_in[2];
    const float* bn0_g  = (const float*)d_in[3];
    const float* bn0_b  = (const float*)d_in[4];
    const float* ese_w  = (const float*)d_in[5];
    const float* ese_b  = (const float*)d_in[6];
    const float* fc1_w  = (const float*)d_in[7];
    const float* fc1_b  = (const float*)d_in[8];
    const float* fc2_w  = (const float*)d_in[9];
    const float* fc2_b  = (const float*)d_in[10];
    const float* f1[3]  = {(const float*)d_in[11], (const float*)d_in[16], (const float*)d_in[21]};
    const float* bvp[3] = {(const float*)d_in[12], (const float*)d_in[17], (const float*)d_in[22]};
    const float* g1[3]  = {(const float*)d_in[13], (const float*)d_in[18], (const float*)d_in[23]};
    const float* be1[3] = {(const float*)d_in[14], (const float*)d_in[19], (const float*)d_in[24]};
    const float* cp[3]  = {(const float*)d_in[15], (const float*)d_in[20], (const float*)d_in[25]};

    // workspace layout (floats)
    float* W = (float*)d_ws;
    size_t off = 0;
    float* F[4];
    const int Cfeat[4] = {16, 48, 80, 112};
    for (int i = 0; i < 4; ++i) { F[i] = W + off; off += (size_t)NB * Cfeat[i] * LSEQ; }
    float* T3 = W + off; off += (size_t)NB * 32 * LSEQ;   // conv(f2) output
    float* T4 = W + off; off += (size_t)NB * 32 * LSEQ;   // conv(feats[i]) output
    float* T1 = W + off; off += (size_t)NB * 80 * LSEQ;   // convT output (<=80 ch)
    float* T2 = W + off; off += (size_t)NB * 80 * LSEQ;   // f2 (<=80 ch)
    float* T5 = W + off; off += (size_t)NB * 112 * LSEQ;  // pre-BN f7 / z-scaled
    float* wt[3];
    const int cinS[3] = {16, 48, 80};
    for (int i = 0; i < 3; ++i) { wt[i] = W + off; off += (size_t)cinS[i] * 32 * 3; }
    float* scale = W + off; off += 112;
    float* shift = W + off; off += 112;
    float* sBuf  = W + off; off += (size_t)NB * 112;
    float* aBuf  = W + off; off += (size_t)NB * 112;
    float* fc1o  = W + off; off += (size_t)NB * 128;

    // ---- stem: WMMA conv + maxpool16, then BN0 (no relu) ----
    stem_conv_pool_wmma<<<dim3(40, NB), 256, 0, stream>>>(x, f0, b0, F[0]);
    bn_stats<<<16, 256, 0, stream>>>(F[0], 16, bn0_g, bn0_b, scale, shift);
    bn_apply<<<eb((size_t)NB * 16 * LSEQ), 256, 0, stream>>>(F[0], F[0], scale, shift, 16, 0);

    // ---- precompute convT weights wt[i] = transpose(flip(f1[i])) ----
    for (int i = 0; i < 3; ++i)
        wtrans<<<eb((size_t)cinS[i] * 32 * 3), 256, 0, stream>>>(f1[i], wt[i], cinS[i]);

    // ---- initial ISTA forward ----
    for (int i = 0; i < 3; ++i) {
        const int cin = cinS[i], C = cin + 32;
        conv_k3_wmma<<<(NB * 20 * 2) / 8, 256, 0, stream>>>(F[i], cin, 0, cin, f1[i], 32, T3);
        build_f3<<<eb((size_t)NB * C * LSEQ), 256, 0, stream>>>(F[i + 1], F[i], T3, bvp[i], cp[i], cin, C);
        bn_stats<<<C, 256, 0, stream>>>(F[i + 1], C, g1[i], be1[i], scale, shift);
        bn_apply<<<eb((size_t)NB * C * LSEQ), 256, 0, stream>>>(F[i + 1], F[i + 1], scale, shift, C, 1);
    }

    // ---- unfold iterations ----
    for (int u = 0; u < 3; ++u) {
        // backward reconstruction: i = 0, 1
        for (int i = 0; i < 2; ++i) {
            const int j = 2 - i;
            const int cj = cinS[j];          // convT out channels (80 then 48)
            const int Csrc = Cfeat[3 - i];   // 112 then 80
            conv_k3_wmma<<<(NB * 20 * (cj / 16)) / 8, 256, 0, stream>>>(
                F[3 - i], Csrc, Csrc - 32, 32, wt[j], cj, T1);
            add_sliced<<<eb((size_t)NB * cj * LSEQ), 256, 0, stream>>>(F[2 - i], T1, F[3 - i], Csrc, cj);
        }
        // forward refinement: i = 0..2
        for (int i = 0; i < 3; ++i) {
            const int cin = cinS[i], C = cin + 32;
            conv_k3_wmma<<<(NB * 20 * (cin / 16)) / 8, 256, 0, stream>>>(
                F[i + 1], C, cin, 32, wt[i], cin, T1);                       // convT last-32
            add_sliced<<<eb((size_t)NB * cin * LSEQ), 256, 0, stream>>>(T2, T1, F[i + 1], C, cin); // f2
            conv_k3_wmma<<<(NB * 20 * 2) / 8, 256, 0, stream>>>(T2, cin, 0, cin, f1[i], 32, T3);   // conv(f2)
            conv_k3_wmma<<<(NB * 20 * 2) / 8, 256, 0, stream>>>(F[i], cin, 0, cin, f1[i], 32, T4); // conv(feats[i])
            combine_f7<<<eb((size_t)NB * C * LSEQ), 256, 0, stream>>>(
                T5, F[i + 1], F[i], T2, T3, T4, bvp[i], cp[i], cin, C);
            bn_stats<<<C, 256, 0, stream>>>(T5, C, g1[i], be1[i], scale, shift);
            bn_apply<<<eb((size_t)NB * C * LSEQ), 256, 0, stream>>>(F[i + 1], T5, scale, shift, C, 1);
        }
    }

    // ---- eSE ----
    ese_mean<<<eb((size_t)NB * 112), 256, 0, stream>>>(F[3], sBuf);
    ese_gate<<<eb((size_t)NB * 112), 256, 0, stream>>>(sBuf, ese_w, ese_b, aBuf);
    ese_scale<<<eb((size_t)NB * 112 * LSEQ), 256, 0, stream>>>(F[3], aBuf, T5);

    // ---- FC1 (WMMA GEMM) + FC2 + log_softmax ----
    fc1_wmma<<<8, 256, 0, stream>>>(T5, fc1_w, fc1_b, fc1o);
    fc2_logsoftmax<<<128, 32, 0, stream>>>(fc1o, fc2_w, fc2_b, (float*)d_out);
}